// HardTripletsMiner_80951543595525
// MI455X (gfx1250) — compile-verified
//
#include <hip/hip_runtime.h>
#include <hip/hip_bf16.h>

typedef __attribute__((ext_vector_type(2))) float v2f;
typedef __attribute__((ext_vector_type(8))) float v8f;
typedef __attribute__((ext_vector_type(4))) int   v4i;

#define N_ROWS 8192
#define DIM 128
#define NCHUNK 8
#define COLS_PER_CHUNK (N_ROWS / NCHUNK)       // 1024
#define TILES_PER_CHUNK (COLS_PER_CHUNK / 16)  // 64
#define LDS_STRIDE 132                         // 128 + 4 pad: bank-conflict-free, 16B aligned rows

// ---------------------------------------------------------------------------
// Async global->LDS staging (CDNA5 GLOBAL_LOAD_ASYNC_TO_LDS_B128, ASYNCcnt).
// The builtin takes typed v4i pointers: (global src, lds dst, imm offset,
// imm cpol); the immediate offset is applied to BOTH addresses (ISA 08 §4.4).
// ---------------------------------------------------------------------------
#if defined(__has_builtin)
#if __has_builtin(__builtin_amdgcn_global_load_async_to_lds_b128)
#define HAVE_ASYNC_LDS_BUILTIN 1
#endif
#if __has_builtin(__builtin_amdgcn_s_wait_asynccnt)
#define HAVE_ASYNCCNT_BUILTIN 1
#endif
#endif

__device__ __forceinline__ void async_stage_32B(const float* src, float* lds_dst) {
#ifdef HAVE_ASYNC_LDS_BUILTIN
  v4i* gp = (v4i*)src;
  v4i* lp = (v4i*)lds_dst;  // generic ptr decayed from __shared__; clang re-derives AS3
  __builtin_amdgcn_global_load_async_to_lds_b128(gp, lp, 0, 0);
  __builtin_amdgcn_global_load_async_to_lds_b128(gp, lp, 16, 0);
#else
  unsigned ldsoff = (unsigned)(unsigned long long)lds_dst;  // low 32 bits of the address give the LDS offset
  asm volatile("global_load_async_to_lds_b128 %0, %1, off" ::"v"(ldsoff), "v"(src) : "memory");
  asm volatile("global_load_async_to_lds_b128 %0, %1, off offset:16" ::"v"(ldsoff), "v"(src) : "memory");
#endif
}

template <int N>
__device__ __forceinline__ void wait_asynccnt_le() {
#ifdef HAVE_ASYNCCNT_BUILTIN
  __builtin_amdgcn_s_wait_asynccnt(N);
#else
  if (N == 0)
    asm volatile("s_wait_asynccnt 0x0" ::: "memory");
  else
    asm volatile("s_wait_asynccnt 0x2" ::: "memory");
#endif
}

// ---------------------------------------------------------------------------
// Kernel 1: row-normalize embeddings; store normalized X and per-row sum(x*x)
// (computed on the *normalized* values, matching the reference numerics).
// One wave32 per row: 32 lanes x float4 = 128 floats.
// ---------------------------------------------------------------------------
__global__ __launch_bounds__(256) void normalize_kernel(const float* __restrict__ E,
                                                        float* __restrict__ Xn,
                                                        float* __restrict__ sq) {
  int lane = threadIdx.x & 31;
  int w    = threadIdx.x >> 5;
  int row  = blockIdx.x * 8 + w;

  float4 e = ((const float4*)(E + (size_t)row * DIM))[lane];
  float s = e.x * e.x + e.y * e.y + e.z * e.z + e.w * e.w;
#pragma unroll
  for (int off = 16; off >= 1; off >>= 1) s += __shfl_xor(s, off, 32);
  float inv = 1.0f / sqrtf(s);
  float4 x = make_float4(e.x * inv, e.y * inv, e.z * inv, e.w * inv);
  float s2 = x.x * x.x + x.y * x.y + x.z * x.z + x.w * x.w;
#pragma unroll
  for (int off = 16; off >= 1; off >>= 1) s2 += __shfl_xor(s2, off, 32);
  ((float4*)(Xn + (size_t)row * DIM))[lane] = x;
  if (lane == 0) sq[row] = s2;
}

// ---------------------------------------------------------------------------
// Kernel 2: fused GEMM (X X^T via V_WMMA_F32_16X16X4_F32) + distance +
// masked hardest-positive argmax / hardest-negative argmin, chunked over
// columns. B tiles staged with async global->LDS, double buffered so the
// next tile's fetch overlaps the current tile's 32 WMMAs.
//
// Wave layout (wave32, per ISA 7.12.2):
//   A 16x4 f32 : lane l holds row (l%16), K = 4*ki + 2*(l/16) + {0,1}
//   B 4x16 f32 : lane l holds col (l%16), same K pattern  (B = X^T row = X row)
//   C 16x16 f32: VGPR v, lane l -> row v + 8*(l/16), col l%16
// ---------------------------------------------------------------------------
__global__ __launch_bounds__(256) void mine_kernel(const float* __restrict__ Xn,
                                                   const float* __restrict__ sq,
                                                   const int* __restrict__ labels,
                                                   float* __restrict__ ppv, int* __restrict__ ppi,
                                                   float* __restrict__ pnv, int* __restrict__ pni,
                                                   int* __restrict__ pflag) {
  __shared__ float Bs[2][16 * LDS_STRIDE];

  const int lane  = threadIdx.x & 31;
  const int w     = threadIdx.x >> 5;
  const int r0    = blockIdx.x * 128 + w * 16;  // this wave's 16-row tile
  const int chunk = blockIdx.y;
  const int cbase = chunk * COLS_PER_CHUNK;

  const int lm   = lane & 15;
  const int hi   = lane >> 4;  // 0 or 1
  const int koff = hi * 2;

  // Per-thread staging coordinates: thread moves 32B of one column-row.
  const int rr = threadIdx.x >> 4;        // 0..15
  const int cc = (threadIdx.x & 15) * 8;  // 0..120

  // Prologue: kick off tile 0 into buffer 0.
  async_stage_32B(Xn + (size_t)(cbase + rr) * DIM + cc, &Bs[0][rr * LDS_STRIDE + cc]);

  // Preload A fragments for the whole K=128 loop: 32 x float2 per lane.
  v2f a[32];
  {
    const v2f* ar = (const v2f*)(Xn + (size_t)(r0 + lm) * DIM + koff);
#pragma unroll
    for (int ki = 0; ki < 32; ++ki) a[ki] = ar[2 * ki];
  }

  // Row-side labels / squared norms for the 8 C-slots this lane owns.
  int labI[8];
  float sqI[8];
#pragma unroll
  for (int v = 0; v < 8; ++v) {
    int r = r0 + v + 8 * hi;
    labI[v] = labels[r];
    sqI[v]  = sq[r];
  }

  float bpv[8], bnv[8];
  int bpi[8], bni[8];
#pragma unroll
  for (int v = 0; v < 8; ++v) {
    bpv[v] = -__builtin_inff(); bpi[v] = 0;
    bnv[v] =  __builtin_inff(); bni[v] = 0;
  }
  int flags = 0;  // bits 2v: anyPos(row v), 2v+1: anyNeg(row v)

  for (int t = 0; t < TILES_PER_CHUNK; ++t) {
    const int c0 = cbase + t * 16;
    const float* cur = &Bs[t & 1][0];

    // Stage tile t+1 into the other buffer (it held tile t-1, which every
    // wave finished reading before the trailing barrier of iteration t-1).
    if (t + 1 < TILES_PER_CHUNK) {
      async_stage_32B(Xn + (size_t)(c0 + 16 + rr) * DIM + cc,
                      &Bs[(t + 1) & 1][rr * LDS_STRIDE + cc]);
      wait_asynccnt_le<2>();  // my tile-t ops drained; t+1 ops stay in flight
    } else {
      wait_asynccnt_le<0>();
    }
    __syncthreads();  // all waves' tile-t data resident in LDS

    const int   labJ = labels[c0 + lm];
    const float sqJ  = sq[c0 + lm];

    v8f c = {};
    {
      const float* brow = cur + lm * LDS_STRIDE + koff;
#pragma unroll
      for (int ki = 0; ki < 32; ++ki) {
        v2f b = ((const v2f*)brow)[2 * ki];
        // 8 args: (neg_a, A, neg_b, B, c_mod, C, reuse_a, reuse_b)
        c = __builtin_amdgcn_wmma_f32_16x16x4_f32(false, a[ki], false, b,
                                                  (short)0, c, false, false);
      }
    }

    const int col = c0 + lm;
#pragma unroll
    for (int v = 0; v < 8; ++v) {
      int row = r0 + v + 8 * hi;
      float d2   = sqI[v] + sqJ - 2.0f * c[v];
      float dist = sqrtf(fmaxf(d2, 0.0f));
      bool same  = (labI[v] == labJ);
      bool diag  = (row == col);
      bool isPos = same && !diag;
      bool isNeg = !same;
      float pv = isPos ? dist : 0.0f;              // matches pos_dist = where(pos, d, 0)
      float nv = isNeg ? dist : __builtin_inff();  // matches neg_dist = where(neg, d, inf)
      if (pv > bpv[v] || (pv == bpv[v] && col < bpi[v])) { bpv[v] = pv; bpi[v] = col; }
      if (nv < bnv[v] || (nv == bnv[v] && col < bni[v])) { bnv[v] = nv; bni[v] = col; }
      flags |= (isPos ? (1 << (2 * v)) : 0) | (isNeg ? (1 << (2 * v + 1)) : 0);
    }
    __syncthreads();  // everyone done reading `cur` before it is re-staged
  }

  // Half-wave (16-lane) reduction: lanes 0..15 cover row v, 16..31 cover v+8.
#pragma unroll
  for (int v = 0; v < 8; ++v) {
    float pvv = bpv[v]; int pii = bpi[v];
    float nvv = bnv[v]; int nii = bni[v];
    int fl = (flags >> (2 * v)) & 3;
#pragma unroll
    for (int off = 8; off >= 1; off >>= 1) {
      float opv = __shfl_xor(pvv, off, 32); int opi = __shfl_xor(pii, off, 32);
      float onv = __shfl_xor(nvv, off, 32); int oni = __shfl_xor(nii, off, 32);
      int   ofl = __shfl_xor(fl, off, 32);
      if (opv > pvv || (opv == pvv && opi < pii)) { pvv = opv; pii = opi; }
      if (onv < nvv || (onv == nvv && oni < nii)) { nvv = onv; nii = oni; }
      fl |= ofl;
    }
    if (lm == 0) {
      int row = r0 + v + 8 * hi;
      int idx = row * NCHUNK + chunk;
      ppv[idx] = pvv; ppi[idx] = pii;
      pnv[idx] = nvv; pni[idx] = nii;
      pflag[idx] = fl;
    }
  }
}

// ---------------------------------------------------------------------------
// Kernel 3: merge the NCHUNK partials per row (ascending column ranges, so
// strictly-better-replace preserves first-occurrence ties) and emit outputs.
// ---------------------------------------------------------------------------
__global__ __launch_bounds__(256) void combine_kernel(const float* __restrict__ ppv,
                                                      const int* __restrict__ ppi,
                                                      const float* __restrict__ pnv,
                                                      const int* __restrict__ pni,
                                                      const int* __restrict__ pflag,
                                                      float* __restrict__ out) {
  int row = blockIdx.x * blockDim.x + threadIdx.x;
  if (row >= N_ROWS) return;
  float bp = -__builtin_inff(); int bpi_ = 0;
  float bn =  __builtin_inff(); int bni_ = 0;
  int fl = 0;
  for (int ck = 0; ck < NCHUNK; ++ck) {
    int idx = row * NCHUNK + ck;
    float pv = ppv[idx]; int pi = ppi[idx];
    float nv = pnv[idx]; int ni = pni[idx];
    if (pv > bp || (pv == bp && pi < bpi_)) { bp = pv; bpi_ = pi; }
    if (nv < bn || (nv == bn && ni < bni_)) { bn = nv; bni_ = ni; }
    fl |= pflag[idx];
  }
  out[row]              = (float)row;    // anchor_inds
  out[N_ROWS + row]     = (float)bpi_;   // hardest_pos_inds
  out[2 * N_ROWS + row] = (float)bni_;   // hardest_neg_inds
  out[3 * N_ROWS + row] = ((fl & 1) && (fl & 2)) ? 1.0f : 0.0f;  // keep
}

extern "C" void kernel_launch(void* const* d_in, const int* in_sizes, int n_in,
                              void* d_out, int out_size, void* d_ws, size_t ws_size,
                              hipStream_t stream) {
  const float* E      = (const float*)d_in[0];
  const int*   labels = (const int*)d_in[1];
  float* out = (float*)d_out;

  // Workspace carve-up (floats): Xn[N*D] | sq[N] | ppv | ppi | pnv | pni | pflag
  float* ws  = (float*)d_ws;
  float* Xn  = ws;
  float* sq  = Xn + (size_t)N_ROWS * DIM;
  float* ppv = sq + N_ROWS;
  int*   ppi = (int*)(ppv + (size_t)N_ROWS * NCHUNK);
  float* pnv = (float*)(ppi + (size_t)N_ROWS * NCHUNK);
  int*   pni = (int*)(pnv + (size_t)N_ROWS * NCHUNK);
  int*   pflag = pni + (size_t)N_ROWS * NCHUNK;

  normalize_kernel<<<N_ROWS / 8, 256, 0, stream>>>(E, Xn, sq);
  dim3 grid(N_ROWS / 128, NCHUNK);
  mine_kernel<<<grid, 256, 0, stream>>>(Xn, sq, labels, ppv, ppi, pnv, pni, pflag);
  combine_kernel<<<(N_ROWS + 255) / 256, 256, 0, stream>>>(ppv, ppi, pnv, pni, pflag, out);
}